// KVLLayer_17239998726563
// MI455X (gfx1250) — compile-verified
//
#include <hip/hip_runtime.h>
#include <math.h>

typedef __attribute__((ext_vector_type(2))) float v2f;
typedef __attribute__((ext_vector_type(8))) float v8f;

// ---------------------------------------------------------------------------
// Kernel 1: precompute angle table  A[b*NB+j] = atan2(s[b*NB+j], c[b*NB+j])
// (12.8M unique atan2 vs 41M in the naive form; entries reuse them 3.2x)
// ---------------------------------------------------------------------------
__global__ void kvl_angle_table(const float* __restrict__ c,
                                const float* __restrict__ s,
                                float* __restrict__ ang, int n4) {
  int i = blockIdx.x * blockDim.x + threadIdx.x;
  if (i >= n4) return;
  float4 cv = ((const float4*)c)[i];
  float4 sv = ((const float4*)s)[i];
  float4 a;
  a.x = atan2f(sv.x, cv.x);
  a.y = atan2f(sv.y, cv.y);
  a.z = atan2f(sv.z, cv.z);
  a.w = atan2f(sv.w, cv.w);
  ((float4*)ang)[i] = a;
}

// ---------------------------------------------------------------------------
// Kernel 2: gather + sign + segmented sum into v_kvl[row*B + b].
// Each thread: one batch b (blockIdx.y), 8 consecutive entries (rows sorted,
// avg 8 entries/row -> register accumulation kills most atomics).
// ---------------------------------------------------------------------------
template <bool HAS_TABLE>
__global__ void kvl_gather_segsum(const float* __restrict__ ang,   // table (HAS_TABLE)
                                  const float* __restrict__ c,     // fallback
                                  const float* __restrict__ s,     // fallback
                                  const float* __restrict__ signs,
                                  const int*   __restrict__ inds,
                                  const int*   __restrict__ rows,
                                  float* __restrict__ vkvl,
                                  int NB, int E, int B) {
  const int tid = blockIdx.x * blockDim.x + threadIdx.x;
  const int b   = blockIdx.y;
  const int e0  = tid * 8;
  if (e0 >= E) return;   // E is a multiple of 8

  const int4   i0 = ((const int4*)(inds + e0))[0];
  const int4   i1 = ((const int4*)(inds + e0))[1];
  const int4   r0 = ((const int4*)(rows + e0))[0];
  const int4   r1 = ((const int4*)(rows + e0))[1];
  const float4 g0 = ((const float4*)(signs + e0))[0];
  const float4 g1 = ((const float4*)(signs + e0))[1];

  const int   idx[8] = {i0.x, i0.y, i0.z, i0.w, i1.x, i1.y, i1.z, i1.w};
  const int   row[8] = {r0.x, r0.y, r0.z, r0.w, r1.x, r1.y, r1.z, r1.w};
  const float sgn[8] = {g0.x, g0.y, g0.z, g0.w, g1.x, g1.y, g1.z, g1.w};

  const float* arow = HAS_TABLE ? (ang + (size_t)b * NB) : nullptr;
  const float* crow = c + (size_t)b * NB;
  const float* srow = s + (size_t)b * NB;

  float acc = 0.0f;
  int   cur = row[0];
#pragma unroll
  for (int k = 0; k < 8; ++k) {
    float a;
    if (HAS_TABLE) {
      a = sgn[k] * arow[idx[k]];
    } else {
      a = atan2f(sgn[k] * srow[idx[k]], crow[idx[k]]);
    }
    if (row[k] != cur) {
      atomicAdd(&vkvl[cur * B + b], acc);
      acc = 0.0f;
      cur = row[k];
    }
    acc += a;
  }
  atomicAdd(&vkvl[cur * B + b], acc);
}

// ---------------------------------------------------------------------------
// Kernel 3: sum(|v|) over NCYC*B floats using V_WMMA_F32_16X16X4_F32.
// A = 16x4 tile of |v| (2 VGPRs/lane, v2f), B = ones(4x16) -> D[m][n] += sum_k A[m,k].
// Sum over all 256 D elements = 16 * sum(A); divide by 16 at the end.
// Control flow is wave-uniform so EXEC is all-ones around the WMMA (required).
// total is a multiple of 64.
// ---------------------------------------------------------------------------
__global__ void kvl_abs_reduce_wmma(const float* __restrict__ v,
                                    float* __restrict__ accum,
                                    long total, int chunksPerWave) {
  const int gtid = blockIdx.x * blockDim.x + threadIdx.x;
  const int wave = gtid >> 5;
  const int lane = gtid & 31;

  v2f ones; ones.x = 1.0f; ones.y = 1.0f;
  v8f acc = {};

  const long base = (long)wave * 64 * chunksPerWave;
  for (int it = 0; it < chunksPerWave; ++it) {
    const long cb = base + (long)it * 64;     // wave-uniform
    if (cb >= total) break;                    // uniform branch: EXEC stays full
    const long off = cb + lane * 2;
    v2f a;
    a.x = fabsf(v[off]);
    a.y = fabsf(v[off + 1]);
    // D = A x ones + C   (fp32, exact)
    acc = __builtin_amdgcn_wmma_f32_16x16x4_f32(
        /*neg_a=*/false, a, /*neg_b=*/false, ones,
        /*c_mod=*/(short)0, acc, /*reuse_a=*/false, /*reuse_b=*/false);
  }

  float part = acc[0] + acc[1] + acc[2] + acc[3] +
               acc[4] + acc[5] + acc[6] + acc[7];
  atomicAdd(accum, part);
}

// ---------------------------------------------------------------------------
// Kernel 4: finalize scalar mean
// ---------------------------------------------------------------------------
__global__ void kvl_finalize(const float* __restrict__ accum,
                             float* __restrict__ out, float scale) {
  if (blockIdx.x == 0 && threadIdx.x == 0) out[0] = accum[0] * scale;
}

// ---------------------------------------------------------------------------
extern "C" void kernel_launch(void* const* d_in, const int* in_sizes, int n_in,
                              void* d_out, int out_size, void* d_ws, size_t ws_size,
                              hipStream_t stream) {
  const float* c     = (const float*)d_in[0];
  const float* s     = (const float*)d_in[1];
  const float* signs = (const float*)d_in[2];
  const int*   inds  = (const int*)d_in[3];
  const int*   rows  = (const int*)d_in[4];
  float*       out   = (float*)d_out;

  const int B    = 256;
  const int BNB  = in_sizes[0];        // B * NB = 12,800,000
  const int NB   = BNB / B;            // 50,000
  const int E    = in_sizes[2];        // 160,000
  const int NCYC = 20000;
  const long NCB = (long)NCYC * B;     // 5,120,000 (multiple of 64)

  // Workspace layout: [accum pad 256B][v_kvl NCB floats][angle table BNB floats]
  char*  ws     = (char*)d_ws;
  float* accum  = (float*)ws;
  float* vkvl   = (float*)(ws + 256);
  float* angtab = (float*)(ws + 256 + NCB * sizeof(float));
  const size_t need_table = 256 + NCB * sizeof(float) + (size_t)BNB * sizeof(float);
  const bool has_table = (ws_size >= need_table);

  hipMemsetAsync(accum, 0, sizeof(float), stream);
  hipMemsetAsync(vkvl, 0, NCB * sizeof(float), stream);

  if (has_table) {
    const int n4 = BNB / 4;                           // 3,200,000
    kvl_angle_table<<<(n4 + 255) / 256, 256, 0, stream>>>(c, s, angtab, n4);
  }

  {
    const int threads_x = E / 8;                      // 20,000
    dim3 grid((threads_x + 255) / 256, B, 1);
    if (has_table)
      kvl_gather_segsum<true><<<grid, 256, 0, stream>>>(angtab, c, s, signs,
                                                        inds, rows, vkvl, NB, E, B);
    else
      kvl_gather_segsum<false><<<grid, 256, 0, stream>>>(nullptr, c, s, signs,
                                                         inds, rows, vkvl, NB, E, B);
  }

  {
    const int blocks = 256;                           // 2048 waves
    const int waves  = blocks * (256 / 32);
    const long chunks = NCB / 64;                     // 80,000
    const int cpw = (int)((chunks + waves - 1) / waves);
    kvl_abs_reduce_wmma<<<blocks, 256, 0, stream>>>(vkvl, accum, NCB, cpw);
  }

  const float scale = 1.0f / (16.0f * (float)NCB);    // /16 for ones-column fanout
  kvl_finalize<<<1, 32, 0, stream>>>(accum, out, scale);
}